// GNNLayer_64269890617631
// MI455X (gfx1250) — compile-verified
//
#include <hip/hip_runtime.h>
#include <cstdint>
#include <cstddef>

// ---------------------------------------------------------------------------
// GNN sparse aggregation, MI455X (gfx1250, wave32).
//   acc[n][b] = sum_{e: dst[e]==n} adj[e]*w[e]*x[b][src[e]]
//   out[b][n] = relu(acc[n][b] * x[0][n]*self_w[n] + bias[n])
// x transposed to [N][64]: per-edge gather = one 256B row, float4 per lane
// over 16 lanes. Accumulator [N][64] is L2-resident (5.12MB << 192MB L2);
// the 82M f32 atomics never leave the chip. Edge metadata staged to LDS via
// CDNA5 async-to-LDS (ASYNCcnt) loads.
// ---------------------------------------------------------------------------

#define TPB 256
#define EPB 256   // edges per block
#define NWAVES 8  // 256/32

// ---- CDNA5 async global->LDS staging (ASYNCcnt path), guarded -------------
#if defined(__has_builtin)
#if __has_builtin(__builtin_amdgcn_global_load_async_to_lds_b32)
#define HAVE_ASYNC_LDS 1
#endif
#if __has_builtin(__builtin_amdgcn_global_load_async_to_lds_b128)
#define HAVE_ASYNC_B128 1
#endif
#endif
#ifndef HAVE_ASYNC_LDS
#define HAVE_ASYNC_LDS 0
#endif
#ifndef HAVE_ASYNC_B128
#define HAVE_ASYNC_B128 0
#endif

#if HAVE_ASYNC_LDS
typedef __attribute__((address_space(1))) int* g_i32_p;  // global scalar
typedef __attribute__((address_space(3))) int* l_i32_p;  // LDS scalar

__device__ __forceinline__ void async_ld_b32(const void* g, void* l) {
  __builtin_amdgcn_global_load_async_to_lds_b32((g_i32_p)g, (l_i32_p)l, 0, 0);
}

#if HAVE_ASYNC_B128
typedef int v4i_gcc __attribute__((vector_size(16)));              // int4
typedef __attribute__((address_space(1))) v4i_gcc* g_v4i_p;        // global
typedef __attribute__((address_space(3))) v4i_gcc* l_v4i_p;        // LDS

__device__ __forceinline__ void async_ld_b128(const void* g, void* l) {
  __builtin_amdgcn_global_load_async_to_lds_b128((g_v4i_p)g, (l_v4i_p)l, 0, 0);
}
#endif

__device__ __forceinline__ void async_wait_all() {
#if defined(__has_builtin) && __has_builtin(__builtin_amdgcn_s_wait_asynccnt)
  __builtin_amdgcn_s_wait_asynccnt(0);
#else
  asm volatile("s_wait_asynccnt 0" ::: "memory");
#endif
}
#endif

__device__ __forceinline__ float fatomic_add(float* p, float v) {
  return __hip_atomic_fetch_add(p, v, __ATOMIC_RELAXED, __HIP_MEMORY_SCOPE_AGENT);
}

// ---------------------------------------------------------------------------
// Kernel 1: x[B,N] -> x_t[N,64] (LDS-tiled transpose) and acc[N,64] = 0
// ---------------------------------------------------------------------------
__global__ void __launch_bounds__(TPB)
k_prep(const float* __restrict__ x, float* __restrict__ x_t,
       float* __restrict__ acc, int N) {
  __shared__ float tile[64][65];  // 65-pitch: bank-conflict-free transpose
  const int n0 = blockIdx.x * 64;
  const int tid = threadIdx.x;

#pragma unroll
  for (int i = 0; i < 16; ++i) {
    int flat = tid + i * TPB;
    int bl = flat >> 6;   // batch 0..63
    int nl = flat & 63;   // node-in-tile (fastest across lanes -> coalesced)
    int n = n0 + nl;
    tile[bl][nl] = (n < N) ? x[(size_t)bl * N + n] : 0.0f;
  }
  __syncthreads();
#pragma unroll
  for (int i = 0; i < 16; ++i) {
    int flat = tid + i * TPB;
    int nl = flat >> 6;
    int bl = flat & 63;   // fastest across lanes -> coalesced [N,64] writes
    int n = n0 + nl;
    if (n < N) {
      size_t o = ((size_t)n << 6) + bl;
      x_t[o] = tile[bl][nl];
      acc[o] = 0.0f;
    }
  }
}

// ---------------------------------------------------------------------------
// Kernel 2: edge scatter. 16 lanes per edge (float4 each); wave32 does 2
// edges per iteration. Metadata staged to LDS via async b128 loads.
// ---------------------------------------------------------------------------
__global__ void __launch_bounds__(TPB)
k_edges(const int* __restrict__ src, const int* __restrict__ dst,
        const float* __restrict__ adj, const float* __restrict__ w,
        const float* __restrict__ x_t, float* __restrict__ acc, int E) {
  __shared__ int   s_src[EPB];
  __shared__ int   s_dst[EPB];
  __shared__ float s_adj[EPB];
  __shared__ float s_w[EPB];

  const int base = blockIdx.x * EPB;
  const int tid = threadIdx.x;

#if HAVE_ASYNC_LDS
  if (base + EPB <= E) {
#if HAVE_ASYNC_B128
    // One async b128 per thread; thread-groups of 64 (wave-uniform) each
    // stage one of the four metadata arrays (64 * 16B = 1KB = EPB elems).
    const int j4 = (tid & 63) << 2;
    const void* g;
    void* l;
    switch (tid >> 6) {
      case 0:  g = src + base + j4; l = &s_src[j4]; break;
      case 1:  g = dst + base + j4; l = &s_dst[j4]; break;
      case 2:  g = adj + base + j4; l = &s_adj[j4]; break;
      default: g = w   + base + j4; l = &s_w[j4];   break;
    }
    async_ld_b128(g, l);
#else
    const int e = base + tid;
    async_ld_b32(src + e, &s_src[tid]);
    async_ld_b32(dst + e, &s_dst[tid]);
    async_ld_b32(adj + e, &s_adj[tid]);
    async_ld_b32(w + e,   &s_w[tid]);
#endif
    async_wait_all();  // per-wave ASYNCcnt; barrier below covers all waves
  } else
#endif
  {
    const int e = base + tid;
    if (e < E) {
      s_src[tid] = src[e];
      s_dst[tid] = dst[e];
      s_adj[tid] = adj[e];
      s_w[tid]   = w[e];
    }
  }
  __syncthreads();

  const int lane = tid & 31;
  const int wv = tid >> 5;
  const int half = lane >> 4;        // 0 or 1: which of the 2 edges
  const int q = (lane & 15) << 2;    // batch offset 0,4,...,60
  const int nE = (E - base < EPB) ? (E - base) : EPB;

  for (int i = wv * 2; i < nE; i += 2 * NWAVES) {
    const int ei = i + half;
    if (ei < nE) {
      const int s = s_src[ei];           // LDS reads: 2 addrs per wave
      const int d = s_dst[ei];
      const float c = s_adj[ei] * s_w[ei];
      const float4 xv = *(const float4*)(x_t + ((size_t)s << 6) + q);
      float* ap = acc + ((size_t)d << 6) + q;
      fatomic_add(ap + 0, c * xv.x);     // global_atomic_add_f32, L2-resident
      fatomic_add(ap + 1, c * xv.y);
      fatomic_add(ap + 2, c * xv.z);
      fatomic_add(ap + 3, c * xv.w);
    }
  }
}

// ---------------------------------------------------------------------------
// Kernel 3: out[b][n] = relu(acc[n][b]*scale[n] + bias[n]); tiled transpose
// ---------------------------------------------------------------------------
__global__ void __launch_bounds__(TPB)
k_final(const float* __restrict__ acc, const float* __restrict__ x,
        const float* __restrict__ self_w, const float* __restrict__ bias,
        float* __restrict__ out, int N) {
  __shared__ float tile[64][65];
  __shared__ float s_scale[64];
  __shared__ float s_bias[64];
  const int n0 = blockIdx.x * 64;
  const int tid = threadIdx.x;

  if (tid < 64) {
    int n = n0 + tid;
    s_scale[tid] = (n < N) ? x[n] * self_w[n] : 0.0f;  // self-loop uses x[0,:]
    s_bias[tid]  = (n < N) ? bias[n] : 0.0f;
  }
#pragma unroll
  for (int i = 0; i < 16; ++i) {
    int flat = tid + i * TPB;
    int nl = flat >> 6;
    int bl = flat & 63;   // fastest -> coalesced [N,64] reads
    int n = n0 + nl;
    tile[bl][nl] = (n < N) ? acc[((size_t)n << 6) + bl] : 0.0f;
  }
  __syncthreads();
#pragma unroll
  for (int i = 0; i < 16; ++i) {
    int flat = tid + i * TPB;
    int bl = flat >> 6;
    int nl = flat & 63;   // fastest -> coalesced [B,N] writes
    int n = n0 + nl;
    if (n < N) {
      float v = tile[bl][nl] * s_scale[nl] + s_bias[nl];
      out[(size_t)bl * N + n] = (v > 0.0f) ? v : 0.0f;
    }
  }
}

// ---------------------------------------------------------------------------
extern "C" void kernel_launch(void* const* d_in, const int* in_sizes, int n_in,
                              void* d_out, int out_size, void* d_ws, size_t ws_size,
                              hipStream_t stream) {
  const float* x      = (const float*)d_in[0];  // [B,N]
  const float* adj    = (const float*)d_in[1];  // [E]
  const float* w      = (const float*)d_in[2];  // [E]
  const float* self_w = (const float*)d_in[3];  // [N]
  const float* bias   = (const float*)d_in[4];  // [N]
  const int*   src    = (const int*)d_in[5];    // [E]
  const int*   dst    = (const int*)d_in[6];    // [E]

  const int E = in_sizes[1];
  const int N = in_sizes[3];

  float* x_t = (float*)d_ws;                    // [N,64]  5.12 MB
  float* acc = x_t + (size_t)N * 64;            // [N,64]  5.12 MB
  float* out = (float*)d_out;                   // [B,N]

  const int nTiles = (N + 63) / 64;             // 313
  const int nEdgeBlocks = (E + EPB - 1) / EPB;  // 5000

  k_prep <<<nTiles,      TPB, 0, stream>>>(x, x_t, acc, N);
  k_edges<<<nEdgeBlocks, TPB, 0, stream>>>(src, dst, adj, w, x_t, acc, E);
  k_final<<<nTiles,      TPB, 0, stream>>>(acc, x, self_w, bias, out, N);
}